// GraphConv_52381421142447
// MI455X (gfx1250) — compile-verified
//
#include <hip/hip_runtime.h>
#include <stdint.h>

#define D      64
#define HOPS   3
#define SLOTS  (HOPS + 1)       // 4 embedding slots per node
#define TILE   256              // edges staged per block
#define GROUP  16               // threads per edge (each handles one float4)

// ---------------------------------------------------------------------------
// Kernel 1: initialize d_out. Slot 0 = concatenated embeddings, slots 1..3 = 0.
// Each thread writes one float4 of the per-node 256-float region (64 float4s).
// ---------------------------------------------------------------------------
__global__ __launch_bounds__(256) void gc_init(
    const float* __restrict__ ue, const float* __restrict__ ie,
    float* __restrict__ out, int n_users, long long n_nodes)
{
    long long gid   = (long long)blockIdx.x * blockDim.x + threadIdx.x;
    long long total = n_nodes * (SLOTS * D / 4);   // 64 float4 stores per node
    if (gid >= total) return;
    long long node = gid >> 6;          // gid / 64
    int       q    = (int)(gid & 63);   // which float4 within node's region
    float4 v;
    if (q < (D / 4)) {                  // slot 0: copy embedding
        const float* src = (node < (long long)n_users)
            ? (ue + node * D + q * 4)
            : (ie + (node - n_users) * D + q * 4);
        v = *reinterpret_cast<const float4*>(src);
    } else {                            // slots 1..3: zero (accumulators)
        v = make_float4(0.f, 0.f, 0.f, 0.f);
    }
    reinterpret_cast<float4*>(out)[node * (SLOTS * D / 4) + q] = v;
}

// ---------------------------------------------------------------------------
// Kernel 2: one SpMM hop, in-place in d_out.
//   gather : out[(col*4 + hop-1)*64 ..]   (slot hop-1)
//   scatter: atomicAdd out[(row*4 + hop)*64 ..]  (slot hop)
// Each block stages TILE edges' (row, col, val) into LDS via
// GLOBAL_LOAD_ASYNC_TO_LDS_B32 (ASYNCcnt), then processes them with 16
// threads per edge (float4 per lane -> coalesced 256B rows).
// ---------------------------------------------------------------------------
__global__ __launch_bounds__(256) void gc_spmm(
    const int* __restrict__ adj_row, const int* __restrict__ adj_col,
    const float* __restrict__ adj_val, float* out, int nnz, int hop)
{
    __shared__ int   s_row[TILE];
    __shared__ int   s_col[TILE];
    __shared__ float s_val[TILE];

    const int       tid  = threadIdx.x;
    const long long base = (long long)blockIdx.x * TILE;
    const long long eg   = base + tid;

    if (eg < (long long)nnz) {
        // Low 32 bits of a flat LDS pointer == workgroup-relative LDS offset.
        uint32_t lr = (uint32_t)(uintptr_t)&s_row[tid];
        uint32_t lc = (uint32_t)(uintptr_t)&s_col[tid];
        uint32_t lv = (uint32_t)(uintptr_t)&s_val[tid];
        uint64_t ar = (uint64_t)(uintptr_t)(adj_row + eg);
        uint64_t ac = (uint64_t)(uintptr_t)(adj_col + eg);
        uint64_t av = (uint64_t)(uintptr_t)(adj_val + eg);
        asm volatile("global_load_async_to_lds_b32 %0, %1, off"
                     :: "v"(lr), "v"(ar) : "memory");
        asm volatile("global_load_async_to_lds_b32 %0, %1, off"
                     :: "v"(lc), "v"(ac) : "memory");
        asm volatile("global_load_async_to_lds_b32 %0, %1, off"
                     :: "v"(lv), "v"(av) : "memory");
    } else {
        s_row[tid] = 0;
        s_col[tid] = 0;
        s_val[tid] = 0.0f;
    }
    asm volatile("s_wait_asynccnt 0" ::: "memory");
    __syncthreads();

    const int sub  = tid / GROUP;   // which edge within the current chunk
    const int comp = tid % GROUP;   // which float4 of the 64-float row

    for (int chunk = 0; chunk < TILE / GROUP; ++chunk) {
        // Prefetch next chunk's gather row into the cache hierarchy.
        if (chunk + 1 < TILE / GROUP) {
            int nc = s_col[(chunk + 1) * GROUP + sub];
            __builtin_prefetch(
                out + ((size_t)nc * SLOTS + (hop - 1)) * D + comp * 4, 0, 0);
        }

        const int       el   = chunk * GROUP + sub;   // local edge index
        const long long eidx = base + el;
        const int   r = s_row[el];   // LDS broadcast across the 16-lane group
        const int   c = s_col[el];
        const float w = s_val[el];

        if (eidx < (long long)nnz) {
            const float4 x = *reinterpret_cast<const float4*>(
                out + ((size_t)c * SLOTS + (hop - 1)) * D + comp * 4);
            float* dst = out + ((size_t)r * SLOTS + hop) * D + comp * 4;
            __hip_atomic_fetch_add(dst + 0, w * x.x, __ATOMIC_RELAXED,
                                   __HIP_MEMORY_SCOPE_AGENT);
            __hip_atomic_fetch_add(dst + 1, w * x.y, __ATOMIC_RELAXED,
                                   __HIP_MEMORY_SCOPE_AGENT);
            __hip_atomic_fetch_add(dst + 2, w * x.z, __ATOMIC_RELAXED,
                                   __HIP_MEMORY_SCOPE_AGENT);
            __hip_atomic_fetch_add(dst + 3, w * x.w, __ATOMIC_RELAXED,
                                   __HIP_MEMORY_SCOPE_AGENT);
        }
    }
}

// ---------------------------------------------------------------------------
// Launch: init (slot0 = embeds, slots1-3 = 0), then 3 stream-ordered hops.
// ---------------------------------------------------------------------------
extern "C" void kernel_launch(void* const* d_in, const int* in_sizes, int n_in,
                              void* d_out, int out_size, void* d_ws, size_t ws_size,
                              hipStream_t stream)
{
    const float* ue  = (const float*)d_in[0];
    const float* ie  = (const float*)d_in[1];
    const int*   row = (const int*)d_in[2];
    const int*   col = (const int*)d_in[3];
    const float* val = (const float*)d_in[4];

    const int       n_users = in_sizes[0] / D;
    const int       n_items = in_sizes[1] / D;
    const long long n_nodes = (long long)n_users + (long long)n_items;
    const int       nnz     = in_sizes[2];

    float* out = (float*)d_out;

    const long long init_threads = n_nodes * (SLOTS * D / 4);
    const int       init_blocks  = (int)((init_threads + 255) / 256);
    gc_init<<<init_blocks, 256, 0, stream>>>(ue, ie, out, n_users, n_nodes);

    const int blocks = (nnz + TILE - 1) / TILE;
    for (int h = 1; h <= HOPS; ++h) {
        gc_spmm<<<blocks, 256, 0, stream>>>(row, col, val, out, nnz, h);
    }
}